// AttentionLayer_70042326663882
// MI455X (gfx1250) — compile-verified
//
#include <hip/hip_runtime.h>
#include <hip/hip_bf16.h>
#include <math.h>

// Problem shape (fixed by the reference).
#define NROWS 8192
#define FIN   512
#define FOUT  128
#define NEG_BIG (-9.0e15f)

typedef __attribute__((ext_vector_type(2))) float v2f;
typedef __attribute__((ext_vector_type(4))) float v4f;
typedef __attribute__((ext_vector_type(8))) float v8f;

// D(16x16,f32) = A(16x4,f32) x B(4x16,f32) + C   -- CDNA5 wave32 WMMA
__device__ __forceinline__ v8f wmma_f32(v2f a, v2f b, v8f c) {
  return __builtin_amdgcn_wmma_f32_16x16x4_f32(
      /*neg_a=*/false, a, /*neg_b=*/false, b,
      /*c_mod=*/(short)0, c, /*reuse_a=*/false, /*reuse_b=*/false);
}

// -----------------------------------------------------------------------------
// Kernel 1: HT[f][i] = sum_k X[i][k] * W[k][f]   (i.e. (X@W)^T, stored f-major)
// One wave per 16(f) x 16(i) tile. A = W^T tile (M=features), B = X^T tile.
// -----------------------------------------------------------------------------
__global__ void __launch_bounds__(32)
k_gemm_ht(const float* __restrict__ X, const float* __restrict__ W,
          float* __restrict__ HT) {
  const int i0   = blockIdx.x * 16;     // row block
  const int f0   = blockIdx.y * 16;     // feature block
  const int lane = threadIdx.x & 31;
  const int lid  = lane & 15;
  const int hi   = lane >> 4;
  const int koff = hi * 2;

  v8f acc = {0.f, 0.f, 0.f, 0.f, 0.f, 0.f, 0.f, 0.f};
  const float* xrow = X + (size_t)(i0 + lid) * FIN;

#pragma unroll 4
  for (int k = 0; k < FIN; k += 4) {
    // A[m][kk] = W[k+kk][f0+m], m = lid
    v2f a;
    a.x = W[(size_t)(k + koff)     * FOUT + f0 + lid];
    a.y = W[(size_t)(k + koff + 1) * FOUT + f0 + lid];
    // B[kk][n] = X[i0+n][k+kk], n = lid  -> contiguous float2
    v2f b = *(const v2f*)(xrow + k + koff);
    acc = wmma_f32(a, b, acc);
  }

  // c[r] = D[r + 8*hi][lid] -> HT[(f0 + r + 8*hi)][i0 + lid]
#pragma unroll
  for (int r = 0; r < 8; ++r)
    HT[(size_t)(f0 + r + 8 * hi) * NROWS + i0 + lid] = acc[r];
}

// -----------------------------------------------------------------------------
// Kernel 2: f1[i] = h[i]·a1, f2[i] = h[i]·a2  (coalesced over HT columns)
// -----------------------------------------------------------------------------
__global__ void __launch_bounds__(256)
k_featdots(const float* __restrict__ HT, const float* __restrict__ a,
           float* __restrict__ f1, float* __restrict__ f2) {
  const int i = blockIdx.x * 256 + threadIdx.x;
  float s1 = 0.f, s2 = 0.f;
#pragma unroll 4
  for (int f = 0; f < FOUT; ++f) {
    const float h = HT[(size_t)f * NROWS + i];
    s1 = fmaf(h, a[f], s1);
    s2 = fmaf(h, a[FOUT + f], s2);
  }
  f1[i] = s1;
  f2[i] = s2;
}

// -----------------------------------------------------------------------------
// Kernel 3: per-row online softmax stats over masked e-values.
// One 256-thread block per row; coalesced float4 streaming of the adj row;
// (m,l) pairs merged in LDS. Fully-masked rows degrade gracefully to uniform.
// -----------------------------------------------------------------------------
__global__ void __launch_bounds__(256)
k_rowstats(const float* __restrict__ adj, const float* __restrict__ f1,
           const float* __restrict__ f2, float* __restrict__ mrow,
           float* __restrict__ zrow) {
  __shared__ float sm[256];
  __shared__ float sl[256];
  const int i = blockIdx.x;
  const int t = threadIdx.x;
  const float fi = f1[i];
  const float* arow = adj + (size_t)i * NROWS;

  float m = -3.0e38f;  // finite sentinel: exp(m - anything) underflows to 0
  float l = 0.f;
#pragma unroll
  for (int u = 0; u < 8; ++u) {
    const int j = t * 4 + u * 1024;           // coalesced float4 across lanes
    const v4f av = *(const v4f*)(arow + j);
    const v4f fv = *(const v4f*)(f2 + j);
#pragma unroll
    for (int q = 0; q < 4; ++q) {
      float e = fi + fv[q];
      e = e > 0.f ? e : 0.2f * e;             // leaky_relu(0.2)
      const float v  = av[q] > 0.f ? e : NEG_BIG;
      const float mn = fmaxf(m, v);
      l = l * __expf(m - mn) + __expf(v - mn);
      m = mn;
    }
  }
  sm[t] = m;
  sl[t] = l;
  __syncthreads();
  for (int s = 128; s > 0; s >>= 1) {
    if (t < s) {
      const float m2 = sm[t + s], l2 = sl[t + s];
      const float mn = fmaxf(m, m2);
      l = l * __expf(m - mn) + l2 * __expf(m2 - mn);
      m = mn;
      sm[t] = m;
      sl[t] = l;
    }
    __syncthreads();
  }
  if (t == 0) {
    mrow[i] = m;
    zrow[i] = l;
  }
}

// -----------------------------------------------------------------------------
// Kernel 4: out = elu( softmax(masked e) @ h ).
// Block = 64 threads (2 waves). Each wave owns rows [i0,i0+16) x 64 features
// (4 WMMA N-tiles). K loop over j in chunks of 16: the 16x16 P tile is built
// in registers (exactly the A-fragment elements each lane needs), adjacency
// tile = one 64B line per row; B-fragments are contiguous float2s from HT.
// -----------------------------------------------------------------------------
__global__ void __launch_bounds__(64)
k_attn(const float* __restrict__ adj, const float* __restrict__ HT,
       const float* __restrict__ f1, const float* __restrict__ f2,
       const float* __restrict__ mrow, const float* __restrict__ zrow,
       float* __restrict__ out) {
  const int i0   = blockIdx.x * 16;
  const int wv   = threadIdx.x >> 5;    // 0..1: which feature half
  const int lane = threadIdx.x & 31;
  const int lid  = lane & 15;
  const int hi   = lane >> 4;
  const int koff = hi * 2;

  const float fi = f1[i0 + lid];        // A-fragment row m = lid
  const float mi = mrow[i0 + lid];
  const float* arow = adj + (size_t)(i0 + lid) * NROWS;

  v8f acc[4];
#pragma unroll
  for (int t = 0; t < 4; ++t)
    acc[t] = (v8f){0.f, 0.f, 0.f, 0.f, 0.f, 0.f, 0.f, 0.f};

  for (int j0 = 0; j0 < NROWS; j0 += 16) {
    // Pull the adjacency row ahead (global_prefetch_b8).
    __builtin_prefetch(arow + j0 + 512, 0, 0);

    // P[m][4s+koff+{0,1}] = exp(masked_leaky(f1[m]+f2[k]) - m_row)
    v2f p[4];
#pragma unroll
    for (int s = 0; s < 4; ++s) {
      const int k = j0 + 4 * s + koff;
      const v2f av = *(const v2f*)(arow + k);
      const v2f fv = *(const v2f*)(f2 + k);
      float e0 = fi + fv.x; e0 = e0 > 0.f ? e0 : 0.2f * e0;
      float e1 = fi + fv.y; e1 = e1 > 0.f ? e1 : 0.2f * e1;
      const float v0 = av.x > 0.f ? e0 : NEG_BIG;
      const float v1 = av.y > 0.f ? e1 : NEG_BIG;
      p[s].x = __expf(v0 - mi);
      p[s].y = __expf(v1 - mi);
    }

    // acc[t] += P(16x16) @ H_chunk(16x16) per feature tile (4 K-steps each).
#pragma unroll
    for (int t = 0; t < 4; ++t) {
      const float* hc =
          HT + (size_t)((wv * 4 + t) * 16 + lid) * NROWS + j0 + koff;
#pragma unroll
      for (int s = 0; s < 4; ++s) {
        const v2f b = *(const v2f*)(hc + 4 * s);  // B[kk][n]=h[j0+kk][feat]
        acc[t] = wmma_f32(p[s], b, acc[t]);
      }
    }
  }

  // Epilogue: divide by Z, apply ELU, store (row-major 8192x128 output).
#pragma unroll
  for (int t = 0; t < 4; ++t) {
#pragma unroll
    for (int r = 0; r < 8; ++r) {
      const int irow = i0 + r + 8 * hi;
      const float x = acc[t][r] / zrow[irow];
      out[(size_t)irow * FOUT + (wv * 4 + t) * 16 + lid] =
          x > 0.f ? x : expm1f(x);
    }
  }
}

// -----------------------------------------------------------------------------
extern "C" void kernel_launch(void* const* d_in, const int* in_sizes, int n_in,
                              void* d_out, int out_size, void* d_ws,
                              size_t ws_size, hipStream_t stream) {
  (void)in_sizes; (void)n_in; (void)out_size; (void)ws_size;

  const float* X   = (const float*)d_in[0];  // 8192 x 512
  const float* adj = (const float*)d_in[1];  // 8192 x 8192
  const float* W   = (const float*)d_in[2];  // 512 x 128
  const float* a   = (const float*)d_in[3];  // 256 x 1
  float* out = (float*)d_out;                // 8192 x 128

  // Workspace layout (~4.2 MB): HT | f1 | f2 | mrow | zrow
  float* HT   = (float*)d_ws;                // FOUT * NROWS
  float* f1   = HT + (size_t)FOUT * NROWS;   // NROWS
  float* f2   = f1 + NROWS;                  // NROWS
  float* mrow = f2 + NROWS;                  // NROWS
  float* zrow = mrow + NROWS;                // NROWS

  k_gemm_ht<<<dim3(NROWS / 16, FOUT / 16), 32, 0, stream>>>(X, W, HT);
  k_featdots<<<NROWS / 256, 256, 0, stream>>>(HT, a, f1, f2);
  k_rowstats<<<NROWS, 256, 0, stream>>>(adj, f1, f2, mrow, zrow);
  k_attn<<<NROWS / 16, 64, 0, stream>>>(adj, HT, f1, f2, mrow, zrow, out);
}